// SandwichLayerGCN_11433202942558
// MI455X (gfx1250) — compile-verified
//
#include <hip/hip_runtime.h>
#include <cstdint>

#define DEVINL __device__ __forceinline__

typedef __attribute__((ext_vector_type(2))) float v2f;
typedef __attribute__((ext_vector_type(8))) float v8f;

namespace {
constexpr int   NN   = 30000;
constexpr int   CIN  = 128;
constexpr int   COUT = 32;
constexpr int   TSEQ = 8;
constexpr int   FF   = 256;     // COUT*TSEQ
constexpr int   EE   = 480000;
constexpr float EPS  = 1e-5f;

DEVINL v8f wmma4(v2f a, v2f b, v8f c) {
  // D = A(16x4 f32) * B(4x16 f32) + C(16x16 f32)
  return __builtin_amdgcn_wmma_f32_16x16x4_f32(false, a, false, b, (short)0, c,
                                               false, false);
}
DEVINL float sigmoidf(float x) { return 1.0f / (1.0f + __expf(-x)); }
} // namespace

// ---------------------------------------------------------------- utilities
__global__ void k_zero(float* __restrict__ p, long n) {
  long i = (long)blockIdx.x * blockDim.x + threadIdx.x;
  long stride = (long)gridDim.x * blockDim.x;
  for (; i < n; i += stride) p[i] = 0.0f;
}

__global__ void k_deg(const int* __restrict__ src, const int* __restrict__ dst,
                      float* __restrict__ degO, float* __restrict__ degI) {
  int e = blockIdx.x * blockDim.x + threadIdx.x;
  if (e < EE) {
    atomicAdd(&degO[src[e]], 1.0f);
    atomicAdd(&degI[dst[e]], 1.0f);
  }
}

// BN finalize: scale = g*rsqrt(var+eps), shift = b - mean*scale
__global__ void k_bnfin(const float* __restrict__ sum, const float* __restrict__ sq,
                        const float* __restrict__ g, const float* __restrict__ b,
                        float* __restrict__ scale, float* __restrict__ shift,
                        int C, float invCount) {
  int i = blockIdx.x * blockDim.x + threadIdx.x;
  if (i < C) {
    float m = sum[i] * invCount;
    float v = sq[i] * invCount - m * m;
    float s = g[i] * rsqrtf(v + EPS);
    scale[i] = s;
    shift[i] = b[i] - m * s;
  }
}

// ----------------------------------------------- gated conv1 (Cin=128, k=3)
// Implicit-im2col GEMM: rows r=(n,t) M=240000, K=384 (kk=i*3+k), cols=32.
// One wave = one 16-row tile x full 32 cols (two 16-col WMMA tiles) x 3 weights.
__global__ __launch_bounds__(256) void k_conv1(
    const float* __restrict__ X,
    const float* __restrict__ w1, const float* __restrict__ w2,
    const float* __restrict__ w3,
    const float* __restrict__ b1, const float* __restrict__ b2,
    const float* __restrict__ b3,
    float* __restrict__ x1, float* __restrict__ gsum, float* __restrict__ gsq) {
  __shared__ float sSum[COUT];
  __shared__ float sSq[COUT];
  if (threadIdx.x < COUT) { sSum[threadIdx.x] = 0.f; sSq[threadIdx.x] = 0.f; }
  __syncthreads();

  const int lane = threadIdx.x & 31;
  const int wave = threadIdx.x >> 5;
  const int tile = blockIdx.x * 8 + wave;     // 15000 tiles total
  const int m0   = tile * 16;
  const int l15  = lane & 15;
  const int half = lane >> 4;
  const int r    = m0 + l15;                  // this lane's A row
  const int n    = r >> 3;
  const int t    = r & 7;
  const long xb  = (long)n * (CIN * TSEQ);

  v8f a1c0 = {}, a1c1 = {}, a2c0 = {}, a2c1 = {}, a3c0 = {}, a3c1 = {};
  const int o0 = l15, o1 = l15 + 16;          // B-fragment column per lane

  for (int kk0 = 0; kk0 < CIN * 3; kk0 += 4) {
    const int kkL = kk0 + 2 * half;
    v2f a;
    {
      int kk = kkL;
      int i = kk / 3, k = kk - 3 * i, tt = t + k - 1;
      int ttc = tt < 0 ? 0 : (tt > 7 ? 7 : tt);
      float raw = X[xb + i * 8 + ttc];
      a.x = (tt >= 0 && tt < 8) ? raw : 0.0f;
      kk = kkL + 1;
      i = kk / 3; k = kk - 3 * i; tt = t + k - 1;
      ttc = tt < 0 ? 0 : (tt > 7 ? 7 : tt);
      raw = X[xb + i * 8 + ttc];
      a.y = (tt >= 0 && tt < 8) ? raw : 0.0f;
    }
    v2f f10, f11, f20, f21, f30, f31;
    f10.x = w1[o0 * 384 + kkL]; f10.y = w1[o0 * 384 + kkL + 1];
    f11.x = w1[o1 * 384 + kkL]; f11.y = w1[o1 * 384 + kkL + 1];
    f20.x = w2[o0 * 384 + kkL]; f20.y = w2[o0 * 384 + kkL + 1];
    f21.x = w2[o1 * 384 + kkL]; f21.y = w2[o1 * 384 + kkL + 1];
    f30.x = w3[o0 * 384 + kkL]; f30.y = w3[o0 * 384 + kkL + 1];
    f31.x = w3[o1 * 384 + kkL]; f31.y = w3[o1 * 384 + kkL + 1];
    a1c0 = wmma4(a, f10, a1c0); a1c1 = wmma4(a, f11, a1c1);
    a2c0 = wmma4(a, f20, a2c0); a2c1 = wmma4(a, f21, a2c1);
    a3c0 = wmma4(a, f30, a3c0); a3c1 = wmma4(a, f31, a3c1);
  }

  const float bb10 = b1[o0], bb20 = b2[o0], bb30 = b3[o0];
  const float bb11 = b1[o1], bb21 = b2[o1], bb31 = b3[o1];
  float ls0 = 0.f, lq0 = 0.f, ls1 = 0.f, lq1 = 0.f;
#pragma unroll
  for (int j = 0; j < 8; ++j) {
    const int m = m0 + j + 8 * half;          // C layout: row j (+8 upper half)
    const int nn = m >> 3, tt = m & 7;
    float v = (a1c0[j] + bb10) * sigmoidf(a2c0[j] + bb20) + (a3c0[j] + bb30);
    v = v > 0.f ? v : 0.f;
    x1[(long)nn * FF + o0 * TSEQ + tt] = v;
    ls0 += v; lq0 += v * v;
    v = (a1c1[j] + bb11) * sigmoidf(a2c1[j] + bb21) + (a3c1[j] + bb31);
    v = v > 0.f ? v : 0.f;
    x1[(long)nn * FF + o1 * TSEQ + tt] = v;
    ls1 += v; lq1 += v * v;
  }
  atomicAdd(&sSum[o0], ls0); atomicAdd(&sSq[o0], lq0);
  atomicAdd(&sSum[o1], ls1); atomicAdd(&sSq[o1], lq1);
  __syncthreads();
  if (threadIdx.x < COUT) {
    atomicAdd(&gsum[threadIdx.x], sSum[threadIdx.x]);
    atomicAdd(&gsq[threadIdx.x], sSq[threadIdx.x]);
  }
}

// ----------------------------------------------- residual projection (k=1)
// GEMM rows=(n,t) M=240000, K=128, cols=32.
__global__ __launch_bounds__(256) void k_res(
    const float* __restrict__ X, const float* __restrict__ wr,
    const float* __restrict__ br, float* __restrict__ resid) {
  const int lane = threadIdx.x & 31;
  const int wave = threadIdx.x >> 5;
  const int tile = blockIdx.x * 8 + wave;
  const int m0   = tile * 16;
  const int l15  = lane & 15;
  const int half = lane >> 4;
  const int r    = m0 + l15;
  const int n    = r >> 3;
  const int t    = r & 7;
  const long xb  = (long)n * (CIN * TSEQ);
  const int o0 = l15, o1 = l15 + 16;

  v8f c0 = {}, c1 = {};
  for (int kk0 = 0; kk0 < CIN; kk0 += 4) {
    const int kkL = kk0 + 2 * half;
    v2f a, g0, g1;
    a.x = X[xb + kkL * 8 + t];
    a.y = X[xb + (kkL + 1) * 8 + t];
    g0.x = wr[o0 * CIN + kkL]; g0.y = wr[o0 * CIN + kkL + 1];
    g1.x = wr[o1 * CIN + kkL]; g1.y = wr[o1 * CIN + kkL + 1];
    c0 = wmma4(a, g0, c0);
    c1 = wmma4(a, g1, c1);
  }
  const float bb0 = br[o0], bb1 = br[o1];
#pragma unroll
  for (int j = 0; j < 8; ++j) {
    const int m = m0 + j + 8 * half;
    const int nn = m >> 3, tt = m & 7;
    resid[(long)nn * FF + o0 * TSEQ + tt] = c0[j] + bb0;
    resid[(long)nn * FF + o1 * TSEQ + tt] = c1[j] + bb1;
  }
}

// -------------------------------- BN(conv1) + D_out^-1/2 scaling, vectorized
__global__ void k_xs(const float* __restrict__ x1, const float* __restrict__ scale,
                     const float* __restrict__ shift, const float* __restrict__ degO,
                     float* __restrict__ xs) {
  long tid = (long)blockIdx.x * blockDim.x + threadIdx.x;   // N*64 float4 units
  if (tid >= (long)NN * 64) return;
  const long n = tid >> 6;
  const int  q = (int)(tid & 63);
  const int  c = q >> 1;                                    // channel of this float4
  const float dinv = rsqrtf(fmaxf(degO[n], 1.0f));
  const float s = scale[c] * dinv, sh = shift[c] * dinv;
  float4 v = ((const float4*)(x1 + n * FF))[q];
  float4 o;
  o.x = v.x * s + sh; o.y = v.y * s + sh;
  o.z = v.z * s + sh; o.w = v.w * s + sh;
  ((float4*)(xs + n * FF))[q] = o;
}

// ------------------------------------------------ edge scatter (segment sum)
__global__ __launch_bounds__(256) void k_scatter(
    const int* __restrict__ src, const int* __restrict__ dst,
    const float* __restrict__ xs, float* __restrict__ agg) {
  const int e = blockIdx.x;
  const long s = src[e], d = dst[e];
  const int f = threadIdx.x;
  atomicAdd(&agg[d * FF + f], xs[s * FF + f]);
}

// --------------------------- GCN GEMM: h = (D_in^-1/2 * agg) @ W + b, + BN1 stats
__global__ __launch_bounds__(256) void k_gemm(
    const float* __restrict__ agg, const float* __restrict__ degI,
    const float* __restrict__ W, const float* __restrict__ bias,
    float* __restrict__ h, float* __restrict__ gsum, float* __restrict__ gsq) {
  const int lane = threadIdx.x & 31;
  const int wave = threadIdx.x >> 5;
  const int gw   = blockIdx.x * 8 + wave;     // 30000 wave-tiles (1875 x 16)
  const int rowTile = gw >> 4;
  const int colTile = gw & 15;
  const int l15  = lane & 15;
  const int half = lane >> 4;
  const int r    = rowTile * 16 + l15;
  const int col  = colTile * 16 + l15;
  const float dinv = rsqrtf(fmaxf(degI[r], 1.0f));
  const long arow = (long)r * FF;

  v8f acc = {};
  for (int kk0 = 0; kk0 < FF; kk0 += 4) {
    const int kkL = kk0 + 2 * half;
    v2f a, b;
    a.x = agg[arow + kkL] * dinv;
    a.y = agg[arow + kkL + 1] * dinv;
    b.x = W[(long)kkL * FF + col];
    b.y = W[(long)(kkL + 1) * FF + col];
    acc = wmma4(a, b, acc);
  }
  const float bc = bias[col];
  float ls = 0.f, lq = 0.f;
#pragma unroll
  for (int j = 0; j < 8; ++j) {
    const int row = rowTile * 16 + j + 8 * half;
    const float v = acc[j] + bc;
    h[(long)row * FF + col] = v;
    ls += v; lq += v * v;
  }
  atomicAdd(&gsum[col], ls);
  atomicAdd(&gsq[col], lq);
}

// ----------------------------------------------- gated conv2 (Cin=32, k=3)
// BN1 folded into A loads. K=96 (kk=i*3+k). Output x2 + BN2 stats.
__global__ __launch_bounds__(256) void k_conv2(
    const float* __restrict__ h, const float* __restrict__ scale1,
    const float* __restrict__ shift1,
    const float* __restrict__ w1, const float* __restrict__ w2,
    const float* __restrict__ w3,
    const float* __restrict__ b1, const float* __restrict__ b2,
    const float* __restrict__ b3,
    float* __restrict__ x2, float* __restrict__ gsum, float* __restrict__ gsq) {
  __shared__ float sSum[COUT];
  __shared__ float sSq[COUT];
  if (threadIdx.x < COUT) { sSum[threadIdx.x] = 0.f; sSq[threadIdx.x] = 0.f; }
  __syncthreads();

  const int lane = threadIdx.x & 31;
  const int wave = threadIdx.x >> 5;
  const int tile = blockIdx.x * 8 + wave;
  const int m0   = tile * 16;
  const int l15  = lane & 15;
  const int half = lane >> 4;
  const int r    = m0 + l15;
  const int n    = r >> 3;
  const int t    = r & 7;
  const long hb  = (long)n * FF;
  const int o0 = l15, o1 = l15 + 16;

  v8f a1c0 = {}, a1c1 = {}, a2c0 = {}, a2c1 = {}, a3c0 = {}, a3c1 = {};

  for (int kk0 = 0; kk0 < COUT * 3; kk0 += 4) {
    const int kkL = kk0 + 2 * half;
    v2f a;
    {
      int kk = kkL;
      int i = kk / 3, k = kk - 3 * i, tt = t + k - 1;
      int ttc = tt < 0 ? 0 : (tt > 7 ? 7 : tt);
      int feat = i * 8 + ttc;
      float raw = h[hb + feat] * scale1[feat] + shift1[feat];
      a.x = (tt >= 0 && tt < 8) ? raw : 0.0f;
      kk = kkL + 1;
      i = kk / 3; k = kk - 3 * i; tt = t + k - 1;
      ttc = tt < 0 ? 0 : (tt > 7 ? 7 : tt);
      feat = i * 8 + ttc;
      raw = h[hb + feat] * scale1[feat] + shift1[feat];
      a.y = (tt >= 0 && tt < 8) ? raw : 0.0f;
    }
    v2f f10, f11, f20, f21, f30, f31;
    f10.x = w1[o0 * 96 + kkL]; f10.y = w1[o0 * 96 + kkL + 1];
    f11.x = w1[o1 * 96 + kkL]; f11.y = w1[o1 * 96 + kkL + 1];
    f20.x = w2[o0 * 96 + kkL]; f20.y = w2[o0 * 96 + kkL + 1];
    f21.x = w2[o1 * 96 + kkL]; f21.y = w2[o1 * 96 + kkL + 1];
    f30.x = w3[o0 * 96 + kkL]; f30.y = w3[o0 * 96 + kkL + 1];
    f31.x = w3[o1 * 96 + kkL]; f31.y = w3[o1 * 96 + kkL + 1];
    a1c0 = wmma4(a, f10, a1c0); a1c1 = wmma4(a, f11, a1c1);
    a2c0 = wmma4(a, f20, a2c0); a2c1 = wmma4(a, f21, a2c1);
    a3c0 = wmma4(a, f30, a3c0); a3c1 = wmma4(a, f31, a3c1);
  }

  const float bb10 = b1[o0], bb20 = b2[o0], bb30 = b3[o0];
  const float bb11 = b1[o1], bb21 = b2[o1], bb31 = b3[o1];
  float ls0 = 0.f, lq0 = 0.f, ls1 = 0.f, lq1 = 0.f;
#pragma unroll
  for (int j = 0; j < 8; ++j) {
    const int m = m0 + j + 8 * half;
    const int nn = m >> 3, tt = m & 7;
    float v = (a1c0[j] + bb10) * sigmoidf(a2c0[j] + bb20) + (a3c0[j] + bb30);
    v = v > 0.f ? v : 0.f;
    x2[(long)nn * FF + o0 * TSEQ + tt] = v;
    ls0 += v; lq0 += v * v;
    v = (a1c1[j] + bb11) * sigmoidf(a2c1[j] + bb21) + (a3c1[j] + bb31);
    v = v > 0.f ? v : 0.f;
    x2[(long)nn * FF + o1 * TSEQ + tt] = v;
    ls1 += v; lq1 += v * v;
  }
  atomicAdd(&sSum[o0], ls0); atomicAdd(&sSq[o0], lq0);
  atomicAdd(&sSum[o1], ls1); atomicAdd(&sSq[o1], lq1);
  __syncthreads();
  if (threadIdx.x < COUT) {
    atomicAdd(&gsum[threadIdx.x], sSum[threadIdx.x]);
    atomicAdd(&gsq[threadIdx.x], sSq[threadIdx.x]);
  }
}

// ------------------------------------ final: BN2 + residual + ReLU (float4)
__global__ void k_final(const float* __restrict__ x2, const float* __restrict__ resid,
                        const float* __restrict__ scale2, const float* __restrict__ shift2,
                        float* __restrict__ out) {
  long tid = (long)blockIdx.x * blockDim.x + threadIdx.x;
  if (tid >= (long)NN * 64) return;
  const long n = tid >> 6;
  const int  q = (int)(tid & 63);
  const int  c = q >> 1;
  const float s = scale2[c], sh = shift2[c];
  float4 v = ((const float4*)(x2 + n * FF))[q];
  float4 rr = ((const float4*)(resid + n * FF))[q];
  float4 o;
  o.x = fmaxf(v.x * s + sh + rr.x, 0.f);
  o.y = fmaxf(v.y * s + sh + rr.y, 0.f);
  o.z = fmaxf(v.z * s + sh + rr.z, 0.f);
  o.w = fmaxf(v.w * s + sh + rr.w, 0.f);
  ((float4*)(out + n * FF))[q] = o;
}

// ---------------------------------------------------------------- launcher
extern "C" void kernel_launch(void* const* d_in, const int* in_sizes, int n_in,
                              void* d_out, int out_size, void* d_ws, size_t ws_size,
                              hipStream_t stream) {
  (void)in_sizes; (void)n_in; (void)out_size; (void)ws_size;
  // setup_inputs() dict order
  const float* X     = (const float*)d_in[0];
  const int*   src   = (const int*)d_in[1];
  const int*   dst   = (const int*)d_in[2];
  const float* w_res = (const float*)d_in[3];
  const float* b_res = (const float*)d_in[4];
  const float* c1w1  = (const float*)d_in[5];
  const float* c1w2  = (const float*)d_in[6];
  const float* c1w3  = (const float*)d_in[7];
  const float* c1b1  = (const float*)d_in[8];
  const float* c1b2  = (const float*)d_in[9];
  const float* c1b3  = (const float*)d_in[10];
  const float* c2w1  = (const float*)d_in[11];
  const float* c2w2  = (const float*)d_in[12];
  const float* c2w3  = (const float*)d_in[13];
  const float* c2b1  = (const float*)d_in[14];
  const float* c2b2  = (const float*)d_in[15];
  const float* c2b3  = (const float*)d_in[16];
  const float* gcn_w = (const float*)d_in[17];
  const float* gcn_b = (const float*)d_in[18];
  const float* bn_g  = (const float*)d_in[19];
  const float* bn_b  = (const float*)d_in[20];
  const float* bn1_g = (const float*)d_in[21];
  const float* bn1_b = (const float*)d_in[22];
  const float* bn2_g = (const float*)d_in[23];
  const float* bn2_b = (const float*)d_in[24];
  float* out = (float*)d_out;

  float* ws = (float*)d_ws;
  float* deg_out = ws;
  float* deg_in  = ws + NN;
  float* stats   = ws + 2 * NN;               // 640 floats
  float* bn_sum  = stats;                     // 32
  float* bn_sq   = stats + 32;
  float* bn1_sum = stats + 64;                // 256
  float* bn1_sq  = stats + 320;
  float* bn2_sum = stats + 576;               // 32
  float* bn2_sq  = stats + 608;
  float* ss      = ws + 2 * NN + 640;         // 640 floats of scale/shift
  float* scale0 = ss,        *shift0 = ss + 32;
  float* scale1 = ss + 64,   *shift1 = ss + 320;
  float* scale2 = ss + 576,  *shift2 = ss + 608;
  float* bufH   = ws + 2 * NN + 1280;                 // x1, later h  (N*256)
  float* bufRes = bufH   + (long)NN * FF;             // residual     (N*256)
  float* bufXs  = bufRes + (long)NN * FF;             // xs           (N*256)
  float* bufAgg = bufXs  + (long)NN * FF;             // agg, later x2 (N*256)

  const int tilesNT = (NN * TSEQ) / 16;   // 15000, exact
  const int convBlocks = tilesNT / 8;     // 1875 (8 waves/block)
  const int gemmBlocks = (NN / 16) * 16 / 8;  // 1875*16/8 = 3750

  // 1. zero degrees + stat accumulators + agg
  k_zero<<<256, 256, 0, stream>>>(ws, (long)(2 * NN + 1280));
  k_zero<<<2048, 256, 0, stream>>>(bufAgg, (long)NN * FF);
  // 2. degrees
  k_deg<<<(EE + 255) / 256, 256, 0, stream>>>(src, dst, deg_out, deg_in);
  // 3. gated conv1 (WMMA) + BN stats
  k_conv1<<<convBlocks, 256, 0, stream>>>(X, c1w1, c1w2, c1w3, c1b1, c1b2, c1b3,
                                          bufH, bn_sum, bn_sq);
  // 4. residual projection (WMMA)
  k_res<<<convBlocks, 256, 0, stream>>>(X, w_res, b_res, bufRes);
  // 5. BN finalize (per-channel over N*T)
  k_bnfin<<<1, 32, 0, stream>>>(bn_sum, bn_sq, bn_g, bn_b, scale0, shift0,
                                COUT, 1.0f / (float)(NN * TSEQ));
  // 6. normalize + D_out^-1/2
  k_xs<<<(NN * 64 + 255) / 256, 256, 0, stream>>>(bufH, scale0, shift0, deg_out, bufXs);
  // 7. edge scatter-add
  k_scatter<<<EE, 256, 0, stream>>>(src, dst, bufXs, bufAgg);
  // 8. GCN GEMM (WMMA) + BN1 stats  (h overwrites x1 buffer)
  k_gemm<<<gemmBlocks, 256, 0, stream>>>(bufAgg, deg_in, gcn_w, gcn_b,
                                         bufH, bn1_sum, bn1_sq);
  // 9. BN1 finalize (per-feature over N)
  k_bnfin<<<1, 256, 0, stream>>>(bn1_sum, bn1_sq, bn1_g, bn1_b, scale1, shift1,
                                 FF, 1.0f / (float)NN);
  // 10. gated conv2 (WMMA, BN1 folded in) + BN2 stats (x2 overwrites agg buffer)
  k_conv2<<<convBlocks, 256, 0, stream>>>(bufH, scale1, shift1,
                                          c2w1, c2w2, c2w3, c2b1, c2b2, c2b3,
                                          bufAgg, bn2_sum, bn2_sq);
  // 11. BN2 finalize
  k_bnfin<<<1, 32, 0, stream>>>(bn2_sum, bn2_sq, bn2_g, bn2_b, scale2, shift2,
                                COUT, 1.0f / (float)(NN * TSEQ));
  // 12. BN2 apply + residual + ReLU
  k_final<<<(NN * 64 + 255) / 256, 256, 0, stream>>>(bufAgg, bufRes, scale2, shift2, out);
}